// ProjectionLayer_49563922596676
// MI455X (gfx1250) — compile-verified
//
#include <hip/hip_runtime.h>
#include <hip/hip_bf16.h>

// ---------------------------------------------------------------------------
// Fused projection layer for MI455X (gfx1250, wave32):
//   out1 = relu(words @ sh_w + sh_b)                 [B,S,128]
//   out2 = relu(words @ dp_w[corpus] + dp_b[corpus]) [B,S,128]
//   pred = concat(out1,out2) @ fc_w + fc_b           [B,S,4]
// fp32-exact matrix path: V_WMMA_F32_16X16X4_F32.
// Weights are pre-transposed to [H,E] in the workspace so the B operand of
// each WMMA is a single global_load_b64 per lane (K-contiguous pair).
// ---------------------------------------------------------------------------

#define Bsz 64
#define Ssz 512
#define Esz 768
#define Hsz 128
#define Csz 10
#define Osz 4

#define LDA  776   // padded LDS row stride in floats (bank stagger; 776 % 4 == 0)
#define LDO  264   // padded out-tile row stride in floats

typedef float v2f __attribute__((ext_vector_type(2)));
typedef float v8f __attribute__((ext_vector_type(8)));

// ---------------------------------------------------------------------------
// Weight transpose: [E,H] -> [H,E] for dp_w (10 experts) and sh_w (z == 10).
// Classic 32x32 LDS tile transpose, both directions coalesced.
// ---------------------------------------------------------------------------
__global__ __launch_bounds__(256)
void transpose_weights_kernel(const float* __restrict__ dp_w,  // [C,E,H]
                              const float* __restrict__ sh_w,  // [E,H]
                              float*       __restrict__ dpT,   // [C,H,E]
                              float*       __restrict__ shT)   // [H,E]
{
    __shared__ float t[32][33];
    const int z = blockIdx.z;  // 0..9 experts, 10 = shared
    const float* __restrict__ src = (z < Csz) ? dp_w + (size_t)z * Esz * Hsz : sh_w;
    float*       __restrict__ dst = (z < Csz) ? dpT  + (size_t)z * Hsz * Esz : shT;
    const int e0 = blockIdx.x * 32;
    const int h0 = blockIdx.y * 32;
    const int tx = threadIdx.x;       // 0..31
    const int ty = threadIdx.y;       // 0..7
    #pragma unroll
    for (int j = 0; j < 32; j += 8)
        t[ty + j][tx] = src[(size_t)(e0 + ty + j) * Hsz + (h0 + tx)];
    __syncthreads();
    #pragma unroll
    for (int j = 0; j < 32; j += 8)
        dst[(size_t)(h0 + ty + j) * Esz + (e0 + tx)] = t[tx][ty + j];
}

// ---------------------------------------------------------------------------
// Main fused kernel. TRANS=true reads pre-transposed [H,E] weights (b64 B
// loads); TRANS=false is the fallback reading [E,H] directly (2x b32).
// ---------------------------------------------------------------------------
template <bool TRANS>
__global__ __launch_bounds__(512)
void projection_fused_kernel(const float* __restrict__ words,   // [B,S,E]
                             const int*   __restrict__ corpus,  // [B]
                             const float* __restrict__ dp_w,    // [C,E,H]
                             const float* __restrict__ dp_b,    // [C,H]
                             const float* __restrict__ sh_w,    // [E,H]
                             const float* __restrict__ sh_b,    // [H]
                             const float* __restrict__ fc_w,    // [2H,O]
                             const float* __restrict__ fc_b,    // [O]
                             const float* __restrict__ dpT,     // [C,H,E] or null
                             const float* __restrict__ shT,     // [H,E]   or null
                             float*       __restrict__ out)     // [B,S,O]
{
    __shared__ float smem[16 * LDA];   // 49,664 B; reused for the out tile later

    const int b  = blockIdx.y;
    const int s0 = blockIdx.x * 16;
    const int tid  = threadIdx.x;
    const int wv   = tid >> 5;         // wave id 0..15
    const int lane = tid & 31;
    const int m    = lane & 15;        // M row (A/C layout), also N col (B layout)
    const int kh   = (lane >> 4) * 2;  // K sub-offset 0 or 2

    // ---- Stage A: cooperative load of the 16 x 768 words tile into LDS ----
    {
        const float4* wrow = (const float4*)(words + ((size_t)b * Ssz + s0) * Esz);
        #pragma unroll
        for (int i = 0; i < 6; ++i) {
            int q  = tid + i * 512;        // float4 index, 0..3071
            int r  = q / 192;              // 192 float4 per row of 768 floats
            int cv = q - r * 192;
            float4 v = wrow[r * 192 + cv];
            *(float4*)&smem[r * LDA + cv * 4] = v;   // LDA%4==0 -> 16B aligned
        }
    }
    __syncthreads();

    // ---- Stage B: each wave computes one 16x16 tile of the 256-wide concat ----
    const int  c      = corpus[b];
    const bool domain = (wv >= 8);
    const int  hcol0  = (wv & 7) * 16;           // column tile within H=128
    const int  hcol   = hcol0 + m;               // this lane's N column (0..127)

    const float* __restrict__ bias = domain ? (dp_b + (size_t)c * Hsz) : sh_b;
    // B operand base: transposed -> row of length E per column; else column w/ stride H
    const float* __restrict__ wcol;
    if (TRANS) {
        const float* WT = domain ? (dpT + (size_t)c * Hsz * Esz) : shT;
        wcol = WT + (size_t)hcol * Esz;
    } else {
        const float* W = domain ? (dp_w + (size_t)c * Esz * Hsz) : sh_w;
        wcol = W + hcol;
    }

    v8f acc0 = {};
    v8f acc1 = {};
    const float* __restrict__ arow = &smem[m * LDA];

    for (int k0 = 0; k0 < Esz; k0 += 8) {
        {   // K = k0 .. k0+3; this lane supplies k0+kh, k0+kh+1
            int kk = k0 + kh;
            v2f a = *(const v2f*)&arow[kk];                    // ds_load_b64
            v2f bb;
            if (TRANS) {
                bb = *(const v2f*)&wcol[kk];                   // global_load_b64
            } else {
                bb.x = wcol[(size_t)kk * Hsz];
                bb.y = wcol[(size_t)(kk + 1) * Hsz];
            }
            acc0 = __builtin_amdgcn_wmma_f32_16x16x4_f32(
                       false, a, false, bb, (short)0, acc0, false, false);
        }
        {   // K = k0+4 .. k0+7 (independent accumulator for ILP)
            int kk = k0 + 4 + kh;
            v2f a = *(const v2f*)&arow[kk];
            v2f bb;
            if (TRANS) {
                bb = *(const v2f*)&wcol[kk];
            } else {
                bb.x = wcol[(size_t)kk * Hsz];
                bb.y = wcol[(size_t)(kk + 1) * Hsz];
            }
            acc1 = __builtin_amdgcn_wmma_f32_16x16x4_f32(
                       false, a, false, bb, (short)0, acc1, false, false);
        }
    }

    // bias + relu (N == lane&15 for every accumulator VGPR)
    const float bv = bias[hcol];
    v8f tile;
    #pragma unroll
    for (int r = 0; r < 8; ++r) {
        float x = acc0[r] + acc1[r] + bv;
        tile[r] = x > 0.0f ? x : 0.0f;
    }

    // ---- Stage C: spill tiles to LDS (alias A buffer), then tiny fc head ----
    __syncthreads();   // everyone is done reading the A tile

    {
        const int ccol  = wv * 16 + m;        // concat column 0..255
        const int mbase = (lane >> 4) * 8;    // lanes 16-31 hold M = r+8
        #pragma unroll
        for (int r = 0; r < 8; ++r)
            smem[(r + mbase) * LDO + ccol] = tile[r];
    }
    __syncthreads();

    if (tid < 64) {
        const int row = tid >> 2;
        const int o   = tid & 3;
        float sum = fc_b[o];
        const float* orow = &smem[row * LDO];
        #pragma unroll 8
        for (int h = 0; h < 2 * Hsz; ++h)
            sum = fmaf(orow[h], fc_w[h * Osz + o], sum);
        out[((size_t)b * Ssz + s0 + row) * Osz + o] = sum;
    }
}

extern "C" void kernel_launch(void* const* d_in, const int* in_sizes, int n_in,
                              void* d_out, int out_size, void* d_ws, size_t ws_size,
                              hipStream_t stream) {
    const float* words  = (const float*)d_in[0];
    const int*   corpus = (const int*)  d_in[1];
    const float* dp_w   = (const float*)d_in[2];
    const float* dp_b   = (const float*)d_in[3];
    const float* sh_w   = (const float*)d_in[4];
    const float* sh_b   = (const float*)d_in[5];
    const float* fc_w   = (const float*)d_in[6];
    const float* fc_b   = (const float*)d_in[7];
    float*       out    = (float*)d_out;

    dim3 grid(Ssz / 16, Bsz);   // (32, 64) = 2048 workgroups
    dim3 block(512);            // 16 waves of 32

    const size_t dpT_elems = (size_t)Csz * Hsz * Esz;
    const size_t shT_elems = (size_t)Hsz * Esz;
    const size_t need = (dpT_elems + shT_elems) * sizeof(float);

    if (ws_size >= need) {
        float* dpT = (float*)d_ws;
        float* shT = dpT + dpT_elems;
        transpose_weights_kernel<<<dim3(Esz / 32, Hsz / 32, Csz + 1),
                                   dim3(32, 8), 0, stream>>>(dp_w, sh_w, dpT, shT);
        projection_fused_kernel<true><<<grid, block, 0, stream>>>(
            words, corpus, dp_w, dp_b, sh_w, sh_b, fc_w, fc_b, dpT, shT, out);
    } else {
        projection_fused_kernel<false><<<grid, block, 0, stream>>>(
            words, corpus, dp_w, dp_b, sh_w, sh_b, fc_w, fc_b, nullptr, nullptr, out);
    }
}